// myGAT2_85057532330560
// MI455X (gfx1250) — compile-verified
//
#include <hip/hip_runtime.h>
#include <hip/hip_bf16.h>
#include <math.h>

#define N0V 20000
#define N1V 15000
#define N2V 15000
#define NTOT 50000
#define NE 800000
#define NHEAD 4
#define DHID 64
#define HDV 256
#define NETV 8
#define NCLS 16
#define SLOPE 0.2f
#define ALPHAMIX 0.05f

typedef float v2f __attribute__((ext_vector_type(2)));
typedef float v8f __attribute__((ext_vector_type(8)));

static inline int cdiv_h(long long a, long long b) { return (int)((a + b - 1) / b); }

// ---------------------------------------------------------------------------
// C[M, Nout] = X[M,K] @ W[Nout,K]^T (+ bias) using V_WMMA_F32_16X16X4_F32.
// Each wave computes a 16 x (16*NT) strip: one A fragment feeds NT WMMAs.
// Tail M rows are clamped (not branched): garbage rows only feed unstored
// output rows, so the k-loop is branch-free.
// ---------------------------------------------------------------------------
template <int NT>
__global__ __launch_bounds__(256) void gemm_xwt_wmma(
    const float* __restrict__ X, const float* __restrict__ W,
    const float* __restrict__ bias, float* __restrict__ C,
    int M, int K, int Nout)
{
    const int lane    = threadIdx.x & 31;
    const int waveIb  = threadIdx.x >> 5;
    const int wavesPB = blockDim.x >> 5;
    const int grpN    = Nout / (16 * NT);
    const int tilesM  = (M + 15) >> 4;
    const int total   = tilesM * grpN;
    const int halfSel = lane >> 4;   // 0: lanes 0-15 (K k,k+1), 1: lanes 16-31 (K k+2,k+3)
    const int lane15  = lane & 15;

    for (int tile = blockIdx.x * wavesPB + waveIb; tile < total;
         tile += gridDim.x * wavesPB) {
        const int tm = tile / grpN;
        const int tg = tile - tm * grpN;

        int row = tm * 16 + lane15;
        row = (row < M) ? row : (M - 1);            // clamp: branch-free k-loop
        const float* xrow = X + (size_t)row * K;

        const float* wrow[NT];
#pragma unroll
        for (int t = 0; t < NT; ++t)
            wrow[t] = W + (size_t)(tg * 16 * NT + t * 16 + lane15) * K;

        v8f acc[NT] = {};

        for (int k = 0; k < K; k += 4) {
            const int kk = k + halfSel * 2;
            const v2f a = *(const v2f*)(xrow + kk);
#pragma unroll
            for (int t = 0; t < NT; ++t) {
                const v2f b = *(const v2f*)(wrow[t] + kk);
                acc[t] = __builtin_amdgcn_wmma_f32_16x16x4_f32(
                    false, a, false, b, (short)0, acc[t], false, false);
            }
        }

#pragma unroll
        for (int t = 0; t < NT; ++t) {
            const int col = tg * 16 * NT + t * 16 + lane15;
            const float bval = bias ? bias[col] : 0.0f;
#pragma unroll
            for (int r = 0; r < 8; ++r) {
                const int m = tm * 16 + r + halfSel * 8;
                if (m < M)
                    C[(size_t)m * Nout + col] = acc[t][r] + bval;
            }
        }
    }
}

// ---------------------------------------------------------------------------
// el[n,h] = sum_d feat[n,h*Dh+d]*al[h*Dh+d]; er likewise with ar.
// ---------------------------------------------------------------------------
__global__ void node_scores_k(const float* __restrict__ feat,
                              const float* __restrict__ al,
                              const float* __restrict__ ar,
                              float* __restrict__ el, float* __restrict__ er,
                              int Nn, int Hh, int Dh)
{
    int idx = blockIdx.x * blockDim.x + threadIdx.x;
    if (idx >= Nn * Hh) return;
    int n = idx / Hh, h = idx - n * Hh;
    const float* f  = feat + (size_t)n * (Hh * Dh) + h * Dh;
    const float* av = al + h * Dh;
    const float* bv = ar + h * Dh;
    float sl = 0.f, sr = 0.f;
    for (int d = 0; d < Dh; ++d) { float v = f[d]; sl += v * av[d]; sr += v * bv[d]; }
    el[idx] = sl; er[idx] = sr;
}

// ---------------------------------------------------------------------------
// es[et,h] = sum_row ( dot(etab[et], We[row]) * ae[row] ) over rows of head h.
// rows = Hh*64; launch block = rows threads, grid = NET.
// ---------------------------------------------------------------------------
__global__ void edge_type_scores_k(const float* __restrict__ etab,
                                   const float* __restrict__ We,
                                   const float* __restrict__ ae,
                                   float* __restrict__ es, int rows, int Hh)
{
    __shared__ float red[256];
    const int et = blockIdx.x;
    const int r  = threadIdx.x;
    const float* eb = etab + et * 64;
    const float* w  = We + (size_t)r * 64;
    float sum = 0.f;
    for (int k = 0; k < 64; ++k) sum += eb[k] * w[k];
    red[r] = sum * ae[r];
    __syncthreads();
    if (r < Hh) {
        const int dh = rows / Hh;
        float t = 0.f;
        for (int j = 0; j < dh; ++j) t += red[r * dh + j];
        es[et * Hh + r] = t;
    }
}

// order-preserving float atomic max
__device__ inline void atomicMaxF(float* addr, float val)
{
    if (val >= 0.0f) atomicMax((int*)addr, __float_as_int(val));
    else             atomicMin((unsigned int*)addr, __float_as_uint(val));
}

__global__ void fill_k(float* __restrict__ p, float v, int n)
{
    int i = blockIdx.x * blockDim.x + threadIdx.x;
    if (i < n) p[i] = v;
}

__global__ void edge_logits_max_k(const int* __restrict__ src, const int* __restrict__ dst,
                                  const int* __restrict__ et,
                                  const float* __restrict__ el, const float* __restrict__ er,
                                  const float* __restrict__ es,
                                  float* __restrict__ a, float* __restrict__ m,
                                  int Ee, int Hh)
{
    int idx = blockIdx.x * blockDim.x + threadIdx.x;
    if (idx >= Ee * Hh) return;
    int e = idx / Hh, h = idx - e * Hh;
    float v = el[src[e] * Hh + h] + er[dst[e] * Hh + h] + es[et[e] * Hh + h];
    v = (v > 0.f) ? v : v * SLOPE;
    a[idx] = v;
    atomicMaxF(&m[dst[e] * Hh + h], v);
}

__global__ void edge_exp_sum_k(const int* __restrict__ dst,
                               float* __restrict__ a, const float* __restrict__ m,
                               float* __restrict__ s, int Ee, int Hh)
{
    int idx = blockIdx.x * blockDim.x + threadIdx.x;
    if (idx >= Ee * Hh) return;
    int e = idx / Hh, h = idx - e * Hh;
    float v = expf(a[idx] - m[dst[e] * Hh + h]);
    a[idx] = v;
    atomicAdd(&s[dst[e] * Hh + h], v);
}

__global__ void edge_norm_k(const int* __restrict__ dst,
                            float* __restrict__ a, const float* __restrict__ s,
                            const float* __restrict__ ra, int Ee, int Hh)
{
    int idx = blockIdx.x * blockDim.x + threadIdx.x;
    if (idx >= Ee * Hh) return;
    int e = idx / Hh, h = idx - e * Hh;
    float v = a[idx] / (s[dst[e] * Hh + h] + 1e-9f);
    if (ra) v = v * (1.0f - ALPHAMIX) + ra[idx] * ALPHAMIX;
    a[idx] = v;
}

// ---------------------------------------------------------------------------
// out[dst[e]*F + f] += a[e, f/Dh] * feat[src[e]*F + f].  One wave per edge.
// ---------------------------------------------------------------------------
__global__ void aggregate_k(const float* __restrict__ feat, const float* __restrict__ att,
                            const int* __restrict__ src, const int* __restrict__ dst,
                            float* __restrict__ out, int Ee, int F, int Dh, int Hh)
{
    const int lane = threadIdx.x & 31;
    const int wavesTotal = (gridDim.x * blockDim.x) >> 5;
    for (int e = (blockIdx.x * blockDim.x + threadIdx.x) >> 5; e < Ee; e += wavesTotal) {
        const int sN = src[e], dN = dst[e];
        const float* fs = feat + (size_t)sN * F;
        float* od = out + (size_t)dN * F;
        for (int f = lane; f < F; f += 32) {
            float w = att[(size_t)e * Hh + f / Dh];
            atomicAdd(&od[f], w * fs[f]);
        }
    }
}

__global__ void elu_res_k(const float* __restrict__ out, const float* __restrict__ res,
                          float* __restrict__ dstp, long long n)
{
    long long i = (long long)blockIdx.x * blockDim.x + threadIdx.x;
    if (i >= n) return;
    float v = out[i] + (res ? res[i] : 0.0f);
    dstp[i] = (v > 0.f) ? v : (expf(v) - 1.0f);
}

__global__ void add_k(const float* __restrict__ a, const float* __restrict__ b,
                      float* __restrict__ o, long long n)
{
    long long i = (long long)blockIdx.x * blockDim.x + threadIdx.x;
    if (i < n) o[i] = a[i] + b[i];
}

// ---------------------------------------------------------------------------
static void run_gemm(const float* X, const float* W, const float* bias, float* C,
                     int M, int K, int Nout, hipStream_t stream)
{
    if (Nout % 64 == 0) {
        const int tiles = cdiv_h(M, 16) * (Nout / 64);
        const int blocks = cdiv_h(tiles, 8);   // 8 waves per 256-thread block
        gemm_xwt_wmma<4><<<blocks, 256, 0, stream>>>(X, W, bias, C, M, K, Nout);
    } else {
        const int tiles = cdiv_h(M, 16) * (Nout / 16);
        const int blocks = cdiv_h(tiles, 8);
        gemm_xwt_wmma<1><<<blocks, 256, 0, stream>>>(X, W, bias, C, M, K, Nout);
    }
}

extern "C" void kernel_launch(void* const* d_in, const int* in_sizes, int n_in,
                              void* d_out, int out_size, void* d_ws, size_t ws_size,
                              hipStream_t stream)
{
    const float* x0   = (const float*)d_in[0];
    const float* x1   = (const float*)d_in[1];
    const float* x2   = (const float*)d_in[2];
    const int*   src  = (const int*)d_in[3];
    const int*   dst  = (const int*)d_in[4];
    const int*   ety  = (const int*)d_in[5];
    const float* W0 = (const float*)d_in[6],  *b0 = (const float*)d_in[7];
    const float* W1 = (const float*)d_in[8],  *b1 = (const float*)d_in[9];
    const float* W2 = (const float*)d_in[10], *b2 = (const float*)d_in[11];
    const float* Wg0 = (const float*)d_in[12], *etab0 = (const float*)d_in[13],
               * We0 = (const float*)d_in[14], *al0 = (const float*)d_in[15],
               * ar0 = (const float*)d_in[16], *ae0 = (const float*)d_in[17];
    const float* Wg1 = (const float*)d_in[18], *etab1 = (const float*)d_in[19],
               * We1 = (const float*)d_in[20], *al1 = (const float*)d_in[21],
               * ar1 = (const float*)d_in[22], *ae1 = (const float*)d_in[23];
    const float* Wg2 = (const float*)d_in[24], *etab2 = (const float*)d_in[25],
               * We2 = (const float*)d_in[26], *al2 = (const float*)d_in[27],
               * ar2 = (const float*)d_in[28], *ae2 = (const float*)d_in[29];
    const float* resW2 = (const float*)d_in[30];
    float* outp = (float*)d_out;

    // ---- workspace layout (floats) ----
    const size_t NF = (size_t)NTOT * HDV;     // N*256
    float* f = (float*)d_ws;
    float* bufH    = f;             // N*256 : current layer input h
    float* bufFeat = bufH + NF;     // N*256 : projected features
    float* bufOut  = bufFeat + NF;  // N*256 : aggregation output
    float* el   = bufOut + NF;                       // N*H
    float* er   = el + (size_t)NTOT * NHEAD;         // N*H
    float* mb   = er + (size_t)NTOT * NHEAD;         // N*H
    float* sb   = mb + (size_t)NTOT * NHEAD;         // N*H
    float* a0   = sb + (size_t)NTOT * NHEAD;         // E*H (layer-0 attn / res_attn)
    float* a1   = a0 + (size_t)NE * NHEAD;           // E*H (work attn)
    float* esb  = a1 + (size_t)NE * NHEAD;           // NET*H

    const int EH  = NE * NHEAD;
    const int NHn = NTOT * NHEAD;
    const int aggBlocks = cdiv_h((long long)NE * 32, 256);

    // ---- input projections -> bufH [N,64] ----
    run_gemm(x0, W0, b0, bufH,                 N0V, 128, 64, stream);
    run_gemm(x1, W1, b1, bufH + (size_t)N0V*64, N1V, 64, 64, stream);
    run_gemm(x2, W2, b2, bufH + (size_t)(N0V+N1V)*64, N2V, 32, 64, stream);

    // ================= GAT layer 0 =================
    run_gemm(bufH, Wg0, nullptr, bufFeat, NTOT, 64, HDV, stream);
    node_scores_k<<<cdiv_h(NHn, 256), 256, 0, stream>>>(bufFeat, al0, ar0, el, er, NTOT, NHEAD, DHID);
    edge_type_scores_k<<<NETV, 256, 0, stream>>>(etab0, We0, ae0, esb, 256, NHEAD);
    fill_k<<<cdiv_h(NHn, 256), 256, 0, stream>>>(mb, -1e30f, NHn);
    hipMemsetAsync(sb, 0, (size_t)NHn * sizeof(float), stream);
    edge_logits_max_k<<<cdiv_h(EH, 256), 256, 0, stream>>>(src, dst, ety, el, er, esb, a0, mb, NE, NHEAD);
    edge_exp_sum_k<<<cdiv_h(EH, 256), 256, 0, stream>>>(dst, a0, mb, sb, NE, NHEAD);
    edge_norm_k<<<cdiv_h(EH, 256), 256, 0, stream>>>(dst, a0, sb, nullptr, NE, NHEAD);
    hipMemsetAsync(bufOut, 0, NF * sizeof(float), stream);
    aggregate_k<<<aggBlocks, 256, 0, stream>>>(bufFeat, a0, src, dst, bufOut, NE, HDV, DHID, NHEAD);
    elu_res_k<<<cdiv_h((long long)NF, 256), 256, 0, stream>>>(bufOut, nullptr, bufH, (long long)NF);

    // ================= GAT layer 1 (res_attn = a0, identity residual) =====
    run_gemm(bufH, Wg1, nullptr, bufFeat, NTOT, HDV, HDV, stream);
    node_scores_k<<<cdiv_h(NHn, 256), 256, 0, stream>>>(bufFeat, al1, ar1, el, er, NTOT, NHEAD, DHID);
    edge_type_scores_k<<<NETV, 256, 0, stream>>>(etab1, We1, ae1, esb, 256, NHEAD);
    fill_k<<<cdiv_h(NHn, 256), 256, 0, stream>>>(mb, -1e30f, NHn);
    hipMemsetAsync(sb, 0, (size_t)NHn * sizeof(float), stream);
    edge_logits_max_k<<<cdiv_h(EH, 256), 256, 0, stream>>>(src, dst, ety, el, er, esb, a1, mb, NE, NHEAD);
    edge_exp_sum_k<<<cdiv_h(EH, 256), 256, 0, stream>>>(dst, a1, mb, sb, NE, NHEAD);
    edge_norm_k<<<cdiv_h(EH, 256), 256, 0, stream>>>(dst, a1, sb, a0, NE, NHEAD);
    hipMemsetAsync(bufOut, 0, NF * sizeof(float), stream);
    aggregate_k<<<aggBlocks, 256, 0, stream>>>(bufFeat, a1, src, dst, bufOut, NE, HDV, DHID, NHEAD);
    elu_res_k<<<cdiv_h((long long)NF, 256), 256, 0, stream>>>(bufOut, bufH, bufH, (long long)NF);

    // ================= GAT layer 2 (H=1, out C=16, linear residual) =======
    float* feat2 = bufFeat;                          // N*16
    float* res2  = bufFeat + (size_t)NTOT * NCLS;    // N*16
    run_gemm(bufH, Wg2,   nullptr, feat2, NTOT, HDV, NCLS, stream);
    run_gemm(bufH, resW2, nullptr, res2,  NTOT, HDV, NCLS, stream);
    node_scores_k<<<cdiv_h(NTOT, 256), 256, 0, stream>>>(feat2, al2, ar2, el, er, NTOT, 1, NCLS);
    edge_type_scores_k<<<NETV, 64, 0, stream>>>(etab2, We2, ae2, esb, 64, 1);
    fill_k<<<cdiv_h(NTOT, 256), 256, 0, stream>>>(mb, -1e30f, NTOT);
    hipMemsetAsync(sb, 0, (size_t)NTOT * sizeof(float), stream);
    edge_logits_max_k<<<cdiv_h(NE, 256), 256, 0, stream>>>(src, dst, ety, el, er, esb, a1, mb, NE, 1);
    edge_exp_sum_k<<<cdiv_h(NE, 256), 256, 0, stream>>>(dst, a1, mb, sb, NE, 1);
    edge_norm_k<<<cdiv_h(NE, 256), 256, 0, stream>>>(dst, a1, sb, nullptr, NE, 1);
    hipMemsetAsync(bufOut, 0, (size_t)NTOT * NCLS * sizeof(float), stream);
    aggregate_k<<<aggBlocks, 256, 0, stream>>>(feat2, a1, src, dst, bufOut, NE, NCLS, NCLS, 1);
    add_k<<<cdiv_h((long long)NTOT * NCLS, 256), 256, 0, stream>>>(bufOut, res2, outp, (long long)NTOT * NCLS);
}